// GINEncoder_1151051235810
// MI455X (gfx1250) — compile-verified
//
#include <hip/hip_runtime.h>
#include <hip/hip_bf16.h>

typedef __attribute__((ext_vector_type(2))) float v2f;
typedef __attribute__((ext_vector_type(8))) float v8f;

#define N_GRAPHS 256
#define OUT_DIM  320
#define HID      64
#define BN_EPS   1e-5f

// ---------------------------------------------------------------------------
// Edge scatter-add, 4 features per thread:
//   agg[dst[e]*kin + 4f..4f+3] += h[src[e]*h_stride + h_off + 4f..4f+3]
// float4 gather (global_load_b128) + 4 global f32 atomics. Index loads are
// amortized 4x vs the scalar version; atomic count is intrinsic to COO.
// ---------------------------------------------------------------------------
__global__ __launch_bounds__(256) void gin_agg(
    const float* __restrict__ h, int h_stride, int h_off,
    const int* __restrict__ src, const int* __restrict__ dst,
    float* __restrict__ agg, int kin_log2, long long total_vec)
{
    long long idx = (long long)blockIdx.x * blockDim.x + threadIdx.x;
    if (idx >= total_vec) return;
    const int kv_log2 = kin_log2 - 2;               // vec4 chunks per edge
    int f4 = (int)(idx & ((1 << kv_log2) - 1));
    long long e = idx >> kv_log2;
    int s = src[e];
    int d = dst[e];
    const float4 hv =
        *(const float4*)&h[(long long)s * h_stride + h_off + (f4 << 2)];
    long long base = ((long long)d << kin_log2) + (f4 << 2);
    atomicAdd(&agg[base + 0], hv.x);
    atomicAdd(&agg[base + 1], hv.y);
    atomicAdd(&agg[base + 2], hv.z);
    atomicAdd(&agg[base + 3], hv.w);
}

// ---------------------------------------------------------------------------
// Fused GIN MLP via fp32 WMMA (V_WMMA_F32_16X16X4_F32).
// Block = 256 threads = 8 wave32s; each wave owns one 16x16 output tile of a
// 32(nodes) x 64(features) block tile. N % 32 == 0 so no boundary handling.
//   z   = agg + h                      (float4-staged in LDS, pad 4: stride
//                                       == 4 mod 64 banks -> conflict free)
//   hid = relu(z @ W1 + b1)           (WMMA pass 1, K = KIN)
//   out = relu(hid @ W2 + b2)         (WMMA pass 2, K = 64)
// out written pre-BN into xs (stride 320, column offset xs_off); block also
// reduces per-feature sum / sumsq and atomically accumulates BN statistics.
// ---------------------------------------------------------------------------
template <int KIN>
__global__ __launch_bounds__(256) void gin_mlp_wmma(
    const float* __restrict__ agg,
    const float* __restrict__ hsrc, int h_stride, int h_off,
    const float* __restrict__ W1, const float* __restrict__ b1,
    const float* __restrict__ W2, const float* __restrict__ b2,
    float* __restrict__ xs, int xs_off,
    float* __restrict__ stats)
{
    __shared__ float zT[32][KIN + 4];
    __shared__ float hid[32][HID + 4];
    __shared__ float outT[32][HID + 4];

    const int tid  = threadIdx.x;
    const int lane = tid & 31;
    const int wave = tid >> 5;
    const int rowT = wave >> 2;   // 0..1  (16-row tile)
    const int colT = wave & 3;    // 0..3  (16-col tile)
    const int nodeBase = blockIdx.x * 32;

    // ---- stage z = agg + h into LDS as float4 (uniform trip count)
    for (int idx = tid; idx < 32 * (KIN / 4); idx += 256) {
        int r  = idx / (KIN / 4);
        int c4 = idx - r * (KIN / 4);
        int c  = c4 << 2;
        long long node = nodeBase + r;
        float4 a = *(const float4*)&agg[node * KIN + c];
        float4 b = *(const float4*)&hsrc[node * h_stride + h_off + c];
        float4 z;
        z.x = a.x + b.x; z.y = a.y + b.y; z.z = a.z + b.z; z.w = a.w + b.w;
        *(float4*)&zT[r][c] = z;
    }
    __syncthreads();

    // Per-lane fragment coordinates (CDNA5 16x16x4 fp32 WMMA layout):
    //  A 16x4 : lanes 0-15 -> M=lane,  K={0,1}; lanes 16-31 -> M=lane-16, K={2,3}
    //  B 4x16 : lanes 0-15 -> N=lane,  K={0,1}; lanes 16-31 -> N=lane-16, K={2,3}
    //  C/D    : vgpr v -> M = v + (lane>=16 ? 8 : 0), N = lane % 16
    const int mA    = rowT * 16 + (lane & 15);
    const int nB    = colT * 16 + (lane & 15);
    const int kb    = (lane >> 4) << 1;        // 0 or 2
    const int mBase = rowT * 16 + ((lane >> 4) << 3);

    // ---- GEMM 1: hid = relu(zT @ W1 + b1)
    {
        v8f acc = {};
        for (int k = 0; k < KIN; k += 4) {
            v2f a, b;
            a.x = zT[mA][k + kb];
            a.y = zT[mA][k + kb + 1];
            b.x = W1[(k + kb) * HID + nB];
            b.y = W1[(k + kb + 1) * HID + nB];
            acc = __builtin_amdgcn_wmma_f32_16x16x4_f32(
                false, a, false, b, (short)0, acc, false, false);
        }
        const float bias = b1[nB];
#pragma unroll
        for (int v = 0; v < 8; ++v) {
            float val = acc[v] + bias;
            hid[mBase + v][nB] = val > 0.f ? val : 0.f;
        }
    }
    __syncthreads();

    // ---- GEMM 2: out = relu(hid @ W2 + b2); write straight to xs (pre-BN)
    {
        v8f acc = {};
        for (int k = 0; k < HID; k += 4) {
            v2f a, b;
            a.x = hid[mA][k + kb];
            a.y = hid[mA][k + kb + 1];
            b.x = W2[(k + kb) * HID + nB];
            b.y = W2[(k + kb + 1) * HID + nB];
            acc = __builtin_amdgcn_wmma_f32_16x16x4_f32(
                false, a, false, b, (short)0, acc, false, false);
        }
        const float bias = b2[nB];
#pragma unroll
        for (int v = 0; v < 8; ++v) {
            float val = acc[v] + bias;
            val = val > 0.f ? val : 0.f;
            outT[mBase + v][nB] = val;
            xs[(long long)(nodeBase + mBase + v) * OUT_DIM + xs_off + nB] = val;
        }
    }
    __syncthreads();

    // ---- BN statistics: per-feature sum / sumsq over this block's 32 nodes
    if (tid < HID) {
        float s = 0.f, s2 = 0.f;
#pragma unroll
        for (int r = 0; r < 32; ++r) {
            float v = outT[r][tid];
            s  += v;
            s2 += v * v;
        }
        atomicAdd(&stats[tid], s);
        atomicAdd(&stats[HID + tid], s2);
    }
}

// stats[0:64]=sum, [64:128]=sumsq  ->  [128:192]=mean, [192:256]=rsqrt(var+eps)
__global__ void gin_bnstats(float* __restrict__ stats, float inv_n)
{
    int f = threadIdx.x;
    float mean = stats[f] * inv_n;
    float var  = stats[HID + f] * inv_n - mean * mean;
    stats[128 + f] = mean;
    stats[192 + f] = rsqrtf(var + BN_EPS);
}

// ---------------------------------------------------------------------------
// In-place BN on xs column block + segment-sum pooling.
// graph_index is SORTED, so each thread walks 32 consecutive nodes of one
// feature and flushes one pool atomic per graph-run instead of one per node
// (~32x fewer contended atomics). Branches are wave-uniform: all 32 lanes of
// a wave walk the same node sequence (lane varies the feature only).
// ---------------------------------------------------------------------------
__global__ __launch_bounds__(256) void gin_bnapply(
    float* __restrict__ xs, int xs_off,
    const float* __restrict__ stats,
    const float* __restrict__ gamma, const float* __restrict__ beta,
    const int* __restrict__ gidx,
    float* __restrict__ pool, int n_nodes)
{
    const int tid  = threadIdx.x;
    const int f    = tid & (HID - 1);
    const int rg   = tid >> 6;                       // 0..3: 32-row strip
    const int base = blockIdx.x * 128 + rg * 32;

    const float mean = stats[128 + f];
    const float inv  = stats[192 + f];
    const float ga   = gamma[f];
    const float be   = beta[f];

    float sum = 0.f;
    int cur = -1;
    for (int r = 0; r < 32; ++r) {
        int node = base + r;
        if (node >= n_nodes) break;
        long long p = (long long)node * OUT_DIM + xs_off + f;
        float zn = (xs[p] - mean) * inv * ga + be;
        xs[p] = zn;
        int g = gidx[node];
        if (g != cur) {
            if (cur >= 0)
                atomicAdd(&pool[(long long)cur * OUT_DIM + xs_off + f], sum);
            cur = g;
            sum = 0.f;
        }
        sum += zn;
    }
    if (cur >= 0)
        atomicAdd(&pool[(long long)cur * OUT_DIM + xs_off + f], sum);
}

// ---------------------------------------------------------------------------
extern "C" void kernel_launch(void* const* d_in, const int* in_sizes, int n_in,
                              void* d_out, int out_size, void* d_ws, size_t ws_size,
                              hipStream_t stream)
{
    const float* x    = (const float*)d_in[0];
    const int*   ei   = (const int*)d_in[1];
    const int*   gidx = (const int*)d_in[2];

    const int N = in_sizes[0] / 128;   // 100000
    const int E = in_sizes[1] / 2;     // 1200000
    const int* src = ei;
    const int* dst = ei + E;

    float* pool = (float*)d_out;                       // (256, 320)
    float* xs   = (float*)d_out + N_GRAPHS * OUT_DIM;  // (N, 320)

    float* agg   = (float*)d_ws;                                        // N*128 f32 max
    float* stats = (float*)((char*)d_ws + (size_t)N * 128 * sizeof(float));

    hipMemsetAsync(pool, 0, (size_t)N_GRAPHS * OUT_DIM * sizeof(float), stream);

    const float* h = x;
    int h_stride = 128, h_off = 0;

    for (int l = 0; l < 5; ++l) {
        const int kin      = (l == 0) ? 128 : HID;
        const int kin_log2 = (l == 0) ? 7 : 6;
        const float* W1    = (const float*)d_in[3 + 6 * l + 0];
        const float* b1    = (const float*)d_in[3 + 6 * l + 1];
        const float* W2    = (const float*)d_in[3 + 6 * l + 2];
        const float* b2    = (const float*)d_in[3 + 6 * l + 3];
        const float* gamma = (const float*)d_in[3 + 6 * l + 4];
        const float* beta  = (const float*)d_in[3 + 6 * l + 5];
        const int xs_off   = HID * l;

        hipMemsetAsync(agg, 0, (size_t)N * kin * sizeof(float), stream);
        hipMemsetAsync(stats, 0, 128 * sizeof(float), stream);

        long long total_vec = (long long)E * (kin >> 2);
        int blocks = (int)((total_vec + 255) / 256);
        gin_agg<<<blocks, 256, 0, stream>>>(h, h_stride, h_off, src, dst,
                                            agg, kin_log2, total_vec);

        if (kin == 128)
            gin_mlp_wmma<128><<<N / 32, 256, 0, stream>>>(
                agg, h, h_stride, h_off, W1, b1, W2, b2, xs, xs_off, stats);
        else
            gin_mlp_wmma<HID><<<N / 32, 256, 0, stream>>>(
                agg, h, h_stride, h_off, W1, b1, W2, b2, xs, xs_off, stats);

        gin_bnstats<<<1, HID, 0, stream>>>(stats, 1.0f / (float)N);

        gin_bnapply<<<(N + 127) / 128, 256, 0, stream>>>(
            xs, xs_off, stats, gamma, beta, gidx, pool, N);

        // next layer reads the BN'd features straight out of the xs region
        h = xs; h_stride = OUT_DIM; h_off = xs_off;
    }
}